// TemporalGraphNetwork_88536455840071
// MI455X (gfx1250) — compile-verified
//
#include <hip/hip_runtime.h>
#include <cstdint>

#define N_NODES 100000
#define HID 128
#define MSG 640            // 5 * HID
#define SEQ 262144

typedef unsigned int u32x4 __attribute__((ext_vector_type(4)));
typedef int i32x4 __attribute__((ext_vector_type(4)));
typedef int i32x8 __attribute__((ext_vector_type(8)));

// ---------------------------------------------------------------------------
// TDM: 1-D copy of n f32 elements from global -> LDS (CDNA5 tensor DMA).
// D# built per CDNA5 ISA.
// ---------------------------------------------------------------------------
__device__ __forceinline__ void tdm_load_1d_f32(const void* gptr,
                                                unsigned lds_byte_addr,
                                                unsigned nelem) {
  unsigned long long ga = (unsigned long long)(uintptr_t)gptr;
  u32x4 g0;
  g0.x = 1u;                                       // count=1, is_restore=0
  g0.y = lds_byte_addr;                            // lds_addr [63:32]
  g0.z = (unsigned)(ga & 0xFFFFFFFFu);             // global_addr lo
  g0.w = (unsigned)((ga >> 32) & 0x1FFFFFFu)       // global_addr [56:32]
         | (2u << 30);                             // type = 2 ("image")
  i32x8 g1;
  g1[0] = (int)(2u << 16);                         // data_size = 4 bytes
  g1[1] = (int)((nelem & 0xFFFFu) << 16);          // tensor_dim0[15:0]
  g1[2] = (int)((nelem >> 16) | (1u << 16));       // tensor_dim0[31:16] | tensor_dim1=1
  g1[3] = (int)((nelem & 0xFFFFu) << 16);          // tile_dim0 = nelem
  g1[4] = 0;                                       // tile_dim1=0 (unused), tile_dim2=0
  g1[5] = (int)nelem;                              // tensor_dim0_stride lo
  g1[6] = 0;
  g1[7] = 0;
  i32x4 z4 = {0, 0, 0, 0};
#if __clang_major__ >= 23
  i32x8 z8 = {0, 0, 0, 0, 0, 0, 0, 0};
  __builtin_amdgcn_tensor_load_to_lds(g0, g1, z4, z4, z8, 0);
#else
  __builtin_amdgcn_tensor_load_to_lds(g0, g1, z4, z4, 0);
#endif
}

__device__ __forceinline__ void fadd_atomic(float* p, float v) {
  // no-return fp32 atomic add, agent scope -> global_atomic_add_f32
  __hip_atomic_fetch_add(p, v, __ATOMIC_RELAXED, __HIP_MEMORY_SCOPE_AGENT);
}

__device__ __forceinline__ void atom_add4(float* p, float a, float b, float c,
                                          float d) {
  fadd_atomic(p + 0, a);
  fadd_atomic(p + 1, b);
  fadd_atomic(p + 2, c);
  fadd_atomic(p + 3, d);
}

__device__ __forceinline__ float4 cos4(float dt, float4 w, float4 b) {
  return make_float4(__cosf(fmaf(dt, w.x, b.x)), __cosf(fmaf(dt, w.y, b.y)),
                     __cosf(fmaf(dt, w.z, b.z)), __cosf(fmaf(dt, w.w, b.w)));
}

// ---------------------------------------------------------------------------
// Pass 2 (placed first so the disasm snippet shows it): one wave32 per event;
// gather -> message -> atomic scatter.
// ---------------------------------------------------------------------------
__global__ __launch_bounds__(256) void tgn_scatter(
    const int* __restrict__ etype, const int* __restrict__ src_ids,
    const float* __restrict__ src_mask, const int* __restrict__ dst_ids,
    const float* __restrict__ dst_mask, const float* __restrict__ ee,
    const float* __restrict__ event_mask, const float* __restrict__ ts,
    const float* __restrict__ memory, const float* __restrict__ last_update,
    const float* __restrict__ time_w, const float* __restrict__ time_b,
    float* __restrict__ out, float* __restrict__ cnt) {
  __shared__ float s_twb[2 * HID];

  if (threadIdx.x < 32) {  // wave 0 issues the two TDM transfers
    tdm_load_1d_f32(time_w, (unsigned)(uintptr_t)&s_twb[0], HID);
    tdm_load_1d_f32(time_b, (unsigned)(uintptr_t)&s_twb[HID], HID);
    __builtin_amdgcn_s_wait_tensorcnt(0);
  }
  __syncthreads();

  const int lane = threadIdx.x & 31;
  // Wave id is uniform by construction; force it into an SGPR so all
  // per-event scalar loads can select the SMEM (s_load) path.
  const int wid =
      __builtin_amdgcn_readfirstlane((blockIdx.x << 3) | (threadIdx.x >> 5));
  const int nwaves = gridDim.x << 3;

  const float4 tw = ((const float4*)&s_twb[0])[lane];    // ds_load_b128
  const float4 tb = ((const float4*)&s_twb[HID])[lane];  // ds_load_b128
  const int col = lane << 2;

  for (int e = wid; e < SEQ; e += nwaves) {
    if (e + nwaves < SEQ)  // global_prefetch_b8 next event's embedding row
      __builtin_prefetch(&ee[(size_t)(e + nwaves) * HID + col], 0, 1);

    const int sid = src_ids[e];
    const int did = dst_ids[e];
    const float sm = src_mask[e];
    const float dm = dst_mask[e];
    const float em = event_mask[e];
    const float t = ts[e];
    const float ty = (float)etype[e];
    const float slu = last_update[sid];
    const float dlu = last_update[did];

    const float4 ms = *(const float4*)&memory[(size_t)sid * HID + col];
    const float4 md = *(const float4*)&memory[(size_t)did * HID + col];
    const float4 se = make_float4(ms.x * sm, ms.y * sm, ms.z * sm, ms.w * sm);
    const float4 de = make_float4(md.x * dm, md.y * dm, md.z * dm, md.w * dm);

    // faithful reference quirk: both deltas use dst_mask
    const float4 cs = cos4(t - slu * dm, tw, tb);
    const float4 cd = cos4(t - dlu * dm, tw, tb);
    const float4 ev = *(const float4*)&ee[(size_t)e * HID + col];

    float* osrc = out + (size_t)sid * MSG + col;
    float* odst = out + ((size_t)N_NODES + (size_t)did) * MSG + col;

    // src_messages = [type, src_emb, dst_emb, cos_src, ee] * event_mask
    const float tyem = ty * em;
    atom_add4(osrc + 0 * HID, tyem, tyem, tyem, tyem);
    atom_add4(osrc + 1 * HID, se.x * em, se.y * em, se.z * em, se.w * em);
    atom_add4(osrc + 2 * HID, de.x * em, de.y * em, de.z * em, de.w * em);
    atom_add4(osrc + 3 * HID, cs.x * em, cs.y * em, cs.z * em, cs.w * em);
    atom_add4(osrc + 4 * HID, ev.x * em, ev.y * em, ev.z * em, ev.w * em);

    // dst_messages = [type, dst_emb, src_emb, cos_dst, ee] * dst_mask
    const float tydm = ty * dm;
    atom_add4(odst + 0 * HID, tydm, tydm, tydm, tydm);
    atom_add4(odst + 1 * HID, de.x * dm, de.y * dm, de.z * dm, de.w * dm);
    atom_add4(odst + 2 * HID, se.x * dm, se.y * dm, se.z * dm, se.w * dm);
    atom_add4(odst + 3 * HID, cd.x * dm, cd.y * dm, cd.z * dm, cd.w * dm);
    atom_add4(odst + 4 * HID, ev.x * dm, ev.y * dm, ev.z * dm, ev.w * dm);

    if (lane == 0) {
      fadd_atomic(&cnt[sid], 1.0f);
      fadd_atomic(&cnt[N_NODES + did], 1.0f);
    }
  }
}

// ---------------------------------------------------------------------------
// Pass 1: zero the 512 MB output and the per-node counts in workspace.
// ---------------------------------------------------------------------------
__global__ __launch_bounds__(256) void tgn_zero(float4* __restrict__ out4,
                                                long n4,
                                                float* __restrict__ cnt,
                                                int ncnt) {
  long i = (long)blockIdx.x * blockDim.x + threadIdx.x;
  long stride = (long)gridDim.x * blockDim.x;
  float4 z = make_float4(0.f, 0.f, 0.f, 0.f);
  for (long k = i; k < n4; k += stride) out4[k] = z;
  for (long k = i; k < ncnt; k += stride) cnt[k] = 0.f;
}

// ---------------------------------------------------------------------------
// Pass 3: divide each node row by max(count, 1).
// ---------------------------------------------------------------------------
__global__ __launch_bounds__(256) void tgn_norm(float4* __restrict__ out4,
                                                const float* __restrict__ cnt) {
  const int n4 = 2 * N_NODES * (MSG / 4);  // 32e6, fits int
  int i = blockIdx.x * blockDim.x + threadIdx.x;
  int stride = gridDim.x * blockDim.x;
  for (int k = i; k < n4; k += stride) {
    int node = k / (MSG / 4);  // flat index over both tables
    float c = cnt[node];
    float s = 1.0f / fmaxf(c, 1.0f);
    float4 v = out4[k];
    v.x *= s; v.y *= s; v.z *= s; v.w *= s;
    out4[k] = v;
  }
}

extern "C" void kernel_launch(void* const* d_in, const int* in_sizes, int n_in,
                              void* d_out, int out_size, void* d_ws,
                              size_t ws_size, hipStream_t stream) {
  const int* etype = (const int*)d_in[0];
  const int* src_ids = (const int*)d_in[1];
  const float* src_mask = (const float*)d_in[2];
  const int* dst_ids = (const int*)d_in[3];
  const float* dst_mask = (const float*)d_in[4];
  const float* ee = (const float*)d_in[5];
  const float* em = (const float*)d_in[6];
  const float* ts = (const float*)d_in[7];
  const float* memory = (const float*)d_in[8];
  const float* last_update = (const float*)d_in[9];
  const float* time_w = (const float*)d_in[10];
  const float* time_b = (const float*)d_in[11];
  float* out = (float*)d_out;
  float* cnt = (float*)d_ws;  // 2*N_NODES floats = 800 KB scratch

  const long n4 = (long)2 * N_NODES * MSG / 4;
  tgn_zero<<<8192, 256, 0, stream>>>((float4*)out, n4, cnt, 2 * N_NODES);
  tgn_scatter<<<2048, 256, 0, stream>>>(etype, src_ids, src_mask, dst_ids,
                                        dst_mask, ee, em, ts, memory,
                                        last_update, time_w, time_b, out, cnt);
  tgn_norm<<<8192, 256, 0, stream>>>((float4*)out, cnt);
}